// RaggedHSTUAttn_84516366450943
// MI455X (gfx1250) — compile-verified
//
#include <hip/hip_runtime.h>

typedef __attribute__((ext_vector_type(2))) float v2f;
typedef __attribute__((ext_vector_type(8))) float v8f;

#if defined(__has_builtin)
# if __has_builtin(__builtin_amdgcn_global_load_async_to_lds_b128)
#  define HSTU_ASYNC 1
# endif
#endif

namespace {
constexpr int B_ = 32;
constexpr int N_ = 1024;
constexpr int H_ = 4;
constexpr int D_ = 128;
constexpr float ALPHA_ = 0.08838834764831843f;
constexpr float INVN_  = 1.0f / 1024.0f;

constexpr int BLOCK_M = 128;   // rows per workgroup
constexpr int WAVE_M  = 16;    // rows per wave (one WMMA M-tile)
constexpr int TILE_J  = 32;    // key/value columns staged in LDS per step
constexpr int KSTR  = 132;     // K tile: [col][dim], float stride per column
constexpr int VPSTR = 288;     // V tile: [col-pair][2*dim+parity], stride per pair (2*128+32)
constexpr int SSTR  = 36;      // per-wave S transpose scratch stride
}

#ifdef HSTU_ASYNC
typedef int hstu_i32x4 __attribute__((vector_size(16)));
typedef __attribute__((address_space(1))) hstu_i32x4 as1_i32x4;
typedef __attribute__((address_space(3))) hstu_i32x4 as3_i32x4;
#define HSTU_ASYNC_CP_B128(gsrc, ldst) \
    __builtin_amdgcn_global_load_async_to_lds_b128((as1_i32x4*)(gsrc), (as3_i32x4*)(ldst), 0, 0)
#endif

static __device__ __forceinline__ void hstu_wait_async0() {
#ifdef HSTU_ASYNC
# if __has_builtin(__builtin_amdgcn_s_wait_asynccnt)
    __builtin_amdgcn_s_wait_asynccnt(0);
# else
    asm volatile("s_wait_asynccnt 0x0" ::: "memory");
# endif
#endif
}

__global__ __launch_bounds__(256) void hstu_attn_kernel(
    const float* __restrict__ q, const float* __restrict__ kk, const float* __restrict__ vv,
    const long long* __restrict__ seq_offsets, const long long* __restrict__ num_targets,
    float* __restrict__ out)
{
    __shared__ float kt[TILE_J * KSTR];            // K tile: [col][dim]
    __shared__ float vt[(TILE_J / 2) * VPSTR];     // V tile: column-pair interleaved
    __shared__ float st[8 * 16 * SSTR];            // per-wave S scratch: [row][col]

    const int mblks = N_ / BLOCK_M;                // 8
    const int blk  = blockIdx.x;
    const int mblk = blk % mblks;
    const int h    = (blk / mblks) % H_;
    const int b    = blk / (mblks * H_);

    const long long s0 = seq_offsets[b];
    const int L   = (int)(seq_offsets[b + 1] - s0);
    const int m0  = mblk * BLOCK_M;
    if (m0 >= L) return;                           // block-uniform exit
    const int cap = L - (int)num_targets[b];

    const int tid  = threadIdx.x;
    const int wave = tid >> 5;
    const int lane = tid & 31;
    const int lh   = lane >> 4;                    // half-lane (K pair select / M+8)
    const int ln   = lane & 15;

    const int wm0 = m0 + wave * WAVE_M;

    // ---- Q fragment, fp32 A-operand layout: lane (ln,lh): qf[c][r] = Q[ln][4c+2lh+r]
    v2f qf[32];
    {
        const int  mrow = wm0 + ln;
        const bool rv   = (mrow < L);
        const float* qp = q + ((long long)(s0 + (rv ? mrow : 0))) * (H_ * D_) + h * D_;
#pragma unroll
        for (int c = 0; c < 32; ++c) {
            v2f t = {0.0f, 0.0f};
            if (rv) t = *(const v2f*)(qp + c * 4 + lh * 2);
            qf[c] = t;
        }
    }

    v8f acc[8];
#pragma unroll
    for (int dt = 0; dt < 8; ++dt) acc[dt] = v8f{0, 0, 0, 0, 0, 0, 0, 0};

    const int  jend = (m0 + BLOCK_M > cap) ? L : (m0 + BLOCK_M);
    const bool wave_rows_valid  = (wm0 < L);
    const bool wave_has_targets = (wm0 + WAVE_M > cap);

    for (int jt = 0; jt < jend; jt += TILE_J) {
        // ---- K tile stage: 32 cols x 128 dims, 16B per lane x4
#ifdef HSTU_ASYNC
#pragma unroll
        for (int it = 0; it < 4; ++it) {
            const int fidx = tid + it * 256;
            const int col  = fidx >> 5;
            const int dg   = fidx & 31;
            int jc = jt + col;                     // clamp OOB (masked later)
            if (jc > L - 1) jc = L - 1;
            const float* gp = kk + ((long long)(s0 + jc)) * (H_ * D_) + h * D_ + 4 * dg;
            HSTU_ASYNC_CP_B128(gp, &kt[col * KSTR + 4 * dg]);
        }
#else
#pragma unroll
        for (int it = 0; it < 4; ++it) {
            const int fidx = tid + it * 256;
            const int col  = fidx >> 5;
            const int dg   = fidx & 31;
            const int jc   = jt + col;
            float4 kv = {0, 0, 0, 0};
            if (jc < L)
                kv = *(const float4*)(kk + ((long long)(s0 + jc)) * (H_ * D_) + h * D_ + 4 * dg);
            *(float4*)(&kt[col * KSTR + 4 * dg]) = kv;
        }
#endif
        // ---- V tile stage, column-pair interleaved: vt[cp*VPSTR + 2*d + (col&1)] = V[col][d]
#pragma unroll
        for (int it = 0; it < 4; ++it) {
            const int u   = tid + it * 256;        // 0..1023
            const int cp  = u >> 6;                // column pair 0..15
            const int dg2 = u & 63;                // dim pair -> dims {2*dg2, 2*dg2+1}
            const int j0  = jt + 2 * cp;
            const int j1  = j0 + 1;
            v2f a = {0.0f, 0.0f}, c = {0.0f, 0.0f};
            if (j0 < L) a = *(const v2f*)(vv + ((long long)(s0 + j0)) * (H_ * D_) + h * D_ + 2 * dg2);
            if (j1 < L) c = *(const v2f*)(vv + ((long long)(s0 + j1)) * (H_ * D_) + h * D_ + 2 * dg2);
            float4 w;
            w.x = a.x; w.y = c.x; w.z = a.y; w.w = c.y;
            *(float4*)(&vt[cp * VPSTR + 4 * dg2]) = w;
        }

        // ---- L2 prefetch of next tile while this one is consumed
        if (jt + TILE_J < jend) {
#pragma unroll
            for (int it = 0; it < 4; ++it) {
                const int fidx = tid + it * 256;
                const int col  = fidx >> 5;
                const int dg   = fidx & 31;
                int jc = jt + TILE_J + col;
                if (jc > L - 1) jc = L - 1;
                const long long base = ((long long)(s0 + jc)) * (H_ * D_) + h * D_ + 4 * dg;
                __builtin_prefetch(kk + base, 0, 3);
                __builtin_prefetch(vv + base, 0, 3);
            }
        }

        hstu_wait_async0();
        __syncthreads();

        const bool active = wave_rows_valid && ((jt < wm0 + WAVE_M) || wave_has_targets);
        if (active) {
            float* sw = &st[wave * 16 * SSTR];

            // ---- S = mask(silu(Q K^T * alpha)/N): 2 x 32 fp32 WMMA, results -> LDS scratch
#pragma unroll
            for (int nt = 0; nt < 2; ++nt) {
                v8f sacc = v8f{0, 0, 0, 0, 0, 0, 0, 0};
#pragma unroll
                for (int c = 0; c < 32; ++c) {
                    // B-operand (4x16): lane (ln,lh) holds K[jt+nt*16+ln][4c+2lh+r]
                    v2f bf = *(const v2f*)(&kt[(nt * 16 + ln) * KSTR + c * 4 + lh * 2]);
                    sacc = __builtin_amdgcn_wmma_f32_16x16x4_f32(
                        false, qf[c], false, bf, (short)0, sacc, false, false);
                }
                const int j = jt + nt * 16 + ln;
#pragma unroll
                for (int r = 0; r < 8; ++r) {
                    const int i = wm0 + r + 8 * lh;                 // C-layout row
                    const bool valid = (j < L) && (i < L) && ((i >= cap) || (j <= i));
                    const float x = sacc[r] * ALPHA_;
                    const float p = valid ? (x * INVN_) / (1.0f + __expf(-x)) : 0.0f;
                    sw[(r + 8 * lh) * SSTR + nt * 16 + ln] = p;
                }
            }
            asm volatile("s_wait_dscnt 0" ::: "memory");            // LDS in-order per wave

            // ---- O += S(16x32) . V(32x128): 8 K-chunks x 8 D-tiles of fp32 WMMA
#pragma unroll
            for (int c2 = 0; c2 < 8; ++c2) {
                // A-operand: lane (ln,lh) holds S[ln][4*c2+2lh+r]
                v2f af = *(const v2f*)(&sw[ln * SSTR + c2 * 4 + lh * 2]);
#pragma unroll
                for (int dt = 0; dt < 8; ++dt) {
                    // B-operand: single b64 from pair-interleaved V tile
                    v2f bf = *(const v2f*)(&vt[(c2 * 2 + lh) * VPSTR + 2 * (dt * 16 + ln)]);
                    acc[dt] = __builtin_amdgcn_wmma_f32_16x16x4_f32(
                        false, af, false, bf, (short)0, acc[dt], false, false);
                }
            }
        }
        __syncthreads();
    }

    // ---- store O (C-layout): row i = wm0 + r + 8*lh, dim = dt*16 + ln
#pragma unroll
    for (int r = 0; r < 8; ++r) {
        const int i = wm0 + r + 8 * lh;
        if (i < L) {
            float* op = out + ((long long)(s0 + i)) * (H_ * D_) + h * D_;
#pragma unroll
            for (int dt = 0; dt < 8; ++dt) op[dt * 16 + ln] = acc[dt][r];
        }
    }
}

extern "C" void kernel_launch(void* const* d_in, const int* in_sizes, int n_in,
                              void* d_out, int out_size, void* d_ws, size_t ws_size,
                              hipStream_t stream) {
    const float*     q  = (const float*)d_in[0];
    const float*     k  = (const float*)d_in[1];
    const float*     v  = (const float*)d_in[2];
    const long long* so = (const long long*)d_in[3];
    const long long* nt = (const long long*)d_in[4];
    float* out = (float*)d_out;

    const int grid = B_ * H_ * (N_ / BLOCK_M);     // 32*4*8 = 1024 workgroups
    hstu_attn_kernel<<<dim3(grid), dim3(256), 0, stream>>>(q, k, v, so, nt, out);
    (void)in_sizes; (void)n_in; (void)out_size; (void)d_ws; (void)ws_size;
}